// SVGPKanPOD_52166672777688
// MI455X (gfx1250) — compile-verified
//
#include <hip/hip_runtime.h>

typedef __attribute__((ext_vector_type(2))) float v2f;
typedef __attribute__((ext_vector_type(8))) float v8f;

#define B_DIM 2048
#define G_DIM 16384
#define L_DIM 16
#define GT 8            // g-tiles per wave (reuse A registers across these)

__global__ __launch_bounds__(256) void svgp_wmma_kernel(
    const float* __restrict__ alpha_mu,
    const float* __restrict__ alpha_var,
    const float* __restrict__ phi_mu,
    const float* __restrict__ phi_var,
    float* __restrict__ out)
{
    const int tid  = threadIdx.x;
    const int lane = tid & 31;
    const int waveId = blockIdx.x * 8 + (tid >> 5);

    const int G_GROUPS = G_DIM / (16 * GT);           // 128
    const int b_tile  = waveId / G_GROUPS;            // 0..127
    const int g_group = waveId % G_GROUPS;            // 0..127
    const int b0    = b_tile * 16;
    const int gbase = g_group * (16 * GT);

    const int lo = lane & 15;
    const int hi = lane >> 4;

    // ---- Load A operands once; reused for all GT g-tiles. ----
    // f32 A-matrix 16x4 layout: lane(0..15)=row M, VGPR0 holds K=4kk+2*hi,
    // VGPR1 holds K=4kk+1+2*hi  -> contiguous float2 per lane per chunk.
    v2f a_mu[4], a_var[4], a_mu2[4];
    {
        const int arow = (b0 + lo) * L_DIM + 2 * hi;
        #pragma unroll
        for (int kk = 0; kk < 4; ++kk) {
            v2f am = *(const v2f*)(alpha_mu  + arow + 4 * kk);
            v2f av = *(const v2f*)(alpha_var + arow + 4 * kk);
            a_mu[kk]  = am;
            a_var[kk] = av;
            a_mu2[kk] = am * am;
        }
    }

    float* outMean = out;
    float* outVar  = out + (size_t)B_DIM * G_DIM;

    #pragma unroll 1
    for (int t = 0; t < GT; ++t) {
        const int g0   = gbase + t * 16;
        const int prow = (g0 + lo) * L_DIM + 2 * hi;

        // Prefetch next g-tile's phi lines (speculative, L2-resident data).
        if (t + 1 < GT) {
            __builtin_prefetch(phi_mu  + prow + 16 * L_DIM, 0, 3);
            __builtin_prefetch(phi_var + prow + 16 * L_DIM, 0, 3);
        }

        // ---- Load B operands (f32 B-matrix 4x16 mirrors A's address form). ----
        v2f b1[4], b2[4], b3[4];
        #pragma unroll
        for (int kk = 0; kk < 4; ++kk) {
            v2f pm = *(const v2f*)(phi_mu  + prow + 4 * kk);
            v2f pv = *(const v2f*)(phi_var + prow + 4 * kk);
            b1[kk] = pm;              // P_mu
            b2[kk] = pm * pm + pv;    // P_mu^2 + P_var  (folds 2 GEMMs into 1)
            b3[kk] = pv;              // P_var
        }

        // ---- 12x V_WMMA_F32_16X16X4_F32: mean (4) + var (8), K=16. ----
        v8f mean = {};
        v8f var  = {};
        #pragma unroll
        for (int kk = 0; kk < 4; ++kk) {
            mean = __builtin_amdgcn_wmma_f32_16x16x4_f32(
                false, a_mu[kk],  false, b1[kk], (short)0, mean, false, false);
            var  = __builtin_amdgcn_wmma_f32_16x16x4_f32(
                false, a_var[kk], false, b2[kk], (short)0, var,  false, false);
            var  = __builtin_amdgcn_wmma_f32_16x16x4_f32(
                false, a_mu2[kk], false, b3[kk], (short)0, var,  false, false);
        }

        // ---- Store: C/D layout => VGPR v, lanes0-15: M=v, lanes16-31: M=v+8.
        // Each store covers two contiguous 64B segments (coalesced).
        const size_t base = (size_t)(b0 + 8 * hi) * G_DIM + (size_t)(g0 + lo);
        #pragma unroll
        for (int v = 0; v < 8; ++v) {
            const size_t idx = base + (size_t)v * G_DIM;
            outMean[idx] = mean[v];
            outVar[idx]  = var[v];
        }
    }
}

extern "C" void kernel_launch(void* const* d_in, const int* in_sizes, int n_in,
                              void* d_out, int out_size, void* d_ws, size_t ws_size,
                              hipStream_t stream) {
    const float* alpha_mu  = (const float*)d_in[0];
    const float* alpha_var = (const float*)d_in[1];
    const float* phi_mu    = (const float*)d_in[2];
    const float* phi_var   = (const float*)d_in[3];
    float* out = (float*)d_out;

    const int waves  = (B_DIM / 16) * (G_DIM / (16 * GT));  // 16384
    const int blocks = waves / 8;                            // 2048
    svgp_wmma_kernel<<<blocks, 256, 0, stream>>>(alpha_mu, alpha_var,
                                                 phi_mu, phi_var, out);
}